// ce_loss_pair_72499047956669
// MI455X (gfx1250) — compile-verified
//
#include <hip/hip_runtime.h>
#include <hip/hip_bf16.h>

typedef __attribute__((ext_vector_type(4))) float f32x4;
typedef __attribute__((ext_vector_type(4))) int   i32x4;
typedef __attribute__((ext_vector_type(2))) float v2f;
typedef __attribute__((ext_vector_type(8))) float v8f;

#define MAXLEN  256
#define MM_TOT  (MAXLEN * MAXLEN)              // 65536
#define BATCH_N 256
#define NPAIRS  ((unsigned)BATCH_N * MM_TOT)   // 16,777,216 (b,mm) pairs
#define NCHUNK  (NPAIRS / 2u)                  // 8,388,608 float4/int4 chunks

static constexpr unsigned NBLOCKS = 2048;
static constexpr unsigned NTHR    = 256;
static constexpr unsigned STRIDE  = NBLOCKS * NTHR;      // 524,288 threads
static constexpr unsigned CPT     = NCHUNK / STRIDE;     // 16 chunks/thread

// ---------------------------------------------------------------------------
// Per-chunk body: chunk c covers pairs g0=2c, g0+1 (same b, same i, j0 even).
// ---------------------------------------------------------------------------
__device__ __forceinline__ void ce_chunk(unsigned c,
                                         const f32x4* __restrict__ pred4,
                                         const i32x4* __restrict__ true4,
                                         const int* __restrict__ dlen,
                                         float& acc)
{
    f32x4 p = __builtin_nontemporal_load(&pred4[c]);   // {p0,p1, p0,p1}
    i32x4 t = __builtin_nontemporal_load(&true4[c]);   // {t0,t1, t0,t1}

    unsigned g0 = c << 1;
    int b  = (int)(g0 >> 16);
    int i  = (int)((g0 >> 8) & 255u);
    int j0 = (int)(g0 & 255u);

    int  dl = dlen[b];                  // 1 KB table: L0/L2 resident
    bool vi = (i < dl);
    bool v0 = vi && (j0     < dl);
    bool v1 = vi && (j0 + 1 < dl);

    float w0a = (v0 && t.x == 1) ? 0.5f : 0.0f;
    float w1a = (v0 && t.y == 1) ? 1.0f : 0.0f;
    float w0b = (v1 && t.z == 1) ? 0.5f : 0.0f;
    float w1b = (v1 && t.w == 1) ? 1.0f : 0.0f;

    acc = fmaf(w0a, __logf(p.x), acc);
    acc = fmaf(w1a, __logf(p.y), acc);
    acc = fmaf(w0b, __logf(p.z), acc);
    acc = fmaf(w1b, __logf(p.w), acc);
}

__device__ __forceinline__ void block_reduce_store(float acc,
                                                   float* __restrict__ partials)
{
    for (int off = 16; off > 0; off >>= 1)
        acc += __shfl_down(acc, off, 32);

    __shared__ float wsum[8];
    int lane = threadIdx.x & 31;
    int wid  = threadIdx.x >> 5;
    if (lane == 0) wsum[wid] = acc;
    __syncthreads();

    if (wid == 0) {
        float v = (lane < (int)(blockDim.x >> 5)) ? wsum[lane] : 0.0f;
        for (int off = 4; off > 0; off >>= 1)
            v += __shfl_down(v, off, 32);
        if (lane == 0) partials[blockIdx.x] = v;   // plain store: deterministic
    }
}

// ---------------------------------------------------------------------------
// Fast path: fixed geometry => compile-time trip count (16) => the compiler
// can unroll (x4) and keep 8 global_load_b128 in flight per wave (4 KB MLP)
// with partial s_wait_loadcnt, instead of load->wait0->use per iteration.
// ---------------------------------------------------------------------------
__global__ __launch_bounds__(NTHR)
void ce_partial_fixed(const int* __restrict__ ytrue,
                      const float* __restrict__ ypred,
                      const int* __restrict__ dlen,
                      float* __restrict__ partials)
{
    const f32x4* __restrict__ pred4 = (const f32x4*)ypred;
    const i32x4* __restrict__ true4 = (const i32x4*)ytrue;

    unsigned tid = blockIdx.x * NTHR + threadIdx.x;
    float acc = 0.0f;
#pragma unroll 4
    for (unsigned k = 0; k < CPT; ++k)
        ce_chunk(tid + k * STRIDE, pred4, true4, dlen, acc);

    block_reduce_store(acc, partials);
}

// Generic grid-stride fallback (only used if workspace is abnormally small).
__global__ __launch_bounds__(256)
void ce_partial_generic(const int* __restrict__ ytrue,
                        const float* __restrict__ ypred,
                        const int* __restrict__ dlen,
                        float* __restrict__ partials)
{
    const f32x4* __restrict__ pred4 = (const f32x4*)ypred;
    const i32x4* __restrict__ true4 = (const i32x4*)ytrue;

    unsigned tid    = blockIdx.x * blockDim.x + threadIdx.x;
    unsigned stride = gridDim.x * blockDim.x;
    float acc = 0.0f;
    for (unsigned c = tid; c < NCHUNK; c += stride)
        ce_chunk(c, pred4, true4, dlen, acc);

    block_reduce_store(acc, partials);
}

// ---------------------------------------------------------------------------
// Pass 2: one wave reduces the block partials. Cross-lane sum via
// V_WMMA_F32_16X16X4_F32 against an all-ones B matrix:
//   A VGPR0 = lane accumulator, VGPR1 = 0  =>  D[m][n] = acc[m] + acc[m+16]
//   sum_r D[r]: lanes 0-15 = sum(acc[0..7]) + sum(acc[16..23])
//               lanes 16-31 = sum(acc[8..15]) + sum(acc[24..31])
//   grand total = s + shfl_xor(s, 16)
// ---------------------------------------------------------------------------
__global__ __launch_bounds__(32)
void ce_finalize_kernel(const float* __restrict__ partials, int n,
                        const int* __restrict__ dlen,
                        float* __restrict__ out)
{
    int lane = threadIdx.x;   // exactly one wave32, EXEC all ones at WMMA

    float a = 0.0f;
    for (int k = lane; k < n; k += 32) a += partials[k];

    v2f A; A.x = a;    A.y = 0.0f;
    v2f B; B.x = 1.0f; B.y = 1.0f;
    v8f C = {};
    v8f D = __builtin_amdgcn_wmma_f32_16x16x4_f32(
        /*neg_a=*/false, A, /*neg_b=*/false, B,
        /*c_mod=*/(short)0, C, /*reuse_a=*/false, /*reuse_b=*/false);

    float s = D[0] + D[1] + D[2] + D[3] + D[4] + D[5] + D[6] + D[7];
    float total = s + __shfl_xor(s, 16, 32);   // replicated on all lanes

    int ds = 0;
    for (int k = lane; k < BATCH_N; k += 32) ds += dlen[k];
    for (int off = 16; off > 0; off >>= 1) ds += __shfl_xor(ds, off, 32);

    if (lane == 0) out[0] = (-total) / (float)ds;
}

extern "C" void kernel_launch(void* const* d_in, const int* in_sizes, int n_in,
                              void* d_out, int out_size, void* d_ws, size_t ws_size,
                              hipStream_t stream)
{
    const int*   y_true   = (const int*)  d_in[0];  // (256, 65536, 2) int32
    const float* y_pred   = (const float*)d_in[1];  // (256, 65536, 2) fp32
    const int*   doc_len  = (const int*)  d_in[2];  // (256,) int32
    float*       out      = (float*)d_out;
    float*       partials = (float*)d_ws;

    if (ws_size >= (size_t)NBLOCKS * sizeof(float)) {
        ce_partial_fixed<<<NBLOCKS, NTHR, 0, stream>>>(y_true, y_pred, doc_len, partials);
        ce_finalize_kernel<<<1, 32, 0, stream>>>(partials, (int)NBLOCKS, doc_len, out);
    } else {
        int nb = (int)(ws_size / sizeof(float));
        if (nb < 1) nb = 1;
        ce_partial_generic<<<nb, 256, 0, stream>>>(y_true, y_pred, doc_len, partials);
        ce_finalize_kernel<<<1, 32, 0, stream>>>(partials, nb, doc_len, out);
    }
}